// EarthMoversDistance_10290741641975
// MI455X (gfx1250) — compile-verified
//
#include <hip/hip_runtime.h>
#include <hip/hip_bf16.h>

// Sinkhorn-Knopp EMD, linear-space reformulation:
//   K = exp(-C/eps); per iter: f = p/(K g), g = q/(K^T f)  (gated per-batch freeze)
//   dist_b = sum_i f[i,b] * ((C.*K) g)[i,b]
// GEMMs use V_WMMA_F32_16X16X4_F32 (full fp32 matrix pipe).

#define BB   32
#define NN   2048
#define MM   2048
#define NM   (NN * MM)
#define EPS_INV 10.0f
#define N_ITERS 50
#define THRESH  1e-3f
#define KSEG 8                 // k-split factor for GEMM parallelism
#define KCHUNK (MM / KSEG)     // 256 k-values per segment

typedef __attribute__((ext_vector_type(2))) float v2f;
typedef __attribute__((ext_vector_type(8))) float v8f;

// ---------------- init kernels ----------------

__global__ void emd_init_kw(const float* __restrict__ C,
                            float* __restrict__ K, float* __restrict__ W) {
  int idx = blockIdx.x * blockDim.x + threadIdx.x;
  if (idx < NM) {
    float c = C[idx];
    float k = __expf(-c * EPS_INV);
    K[idx] = k;
    W[idx] = c * k;       // for final distance: sum f_i (C.*K)_ij g_j
  }
}

__global__ void emd_init_fg(float* __restrict__ F, float* __restrict__ G,
                            float* __restrict__ err, int* __restrict__ done) {
  int idx = blockIdx.x * blockDim.x + threadIdx.x;
  if (idx < NN * BB) F[idx] = 1.0f;          // log_f0 = 0
  if (idx < MM * BB) G[idx] = 1.0f;          // log_g0 = 0
  if (idx < BB) { err[idx] = 1e30f; done[idx] = 0; }
}

// ---------------- GEMM: Y[seg][i][b] = sum_{k in seg} A[i][k] * X[k][b] ----------------
// A row-major [NN][MM]; X [MM][32]; Ypart [KSEG][NN][32]
__global__ void emd_gemm_nt(const float* __restrict__ A, const float* __restrict__ X,
                            float* __restrict__ Ypart) {
  const int lane = threadIdx.x & 31;
  const int wv   = blockIdx.x * (blockDim.x >> 5) + (threadIdx.x >> 5);
  const int seg  = wv & (KSEG - 1);
  const int ct   = (wv >> 3) & 1;        // column tile: b0 = 0 or 16
  const int rt   = wv >> 4;              // row tile 0..127
  const int i0   = rt << 4;
  const int b0   = ct << 4;
  const int hi   = lane >> 4;            // half-wave selects K pair {0,1} vs {2,3}
  const int m    = lane & 15;

  v8f acc = {};
  const float* arow = A + (size_t)(i0 + m) * MM;   // this lane's A row
  const int kbeg = seg * KCHUNK;
  const int kend = kbeg + KCHUNK;
  for (int k = kbeg; k < kend; k += 4) {
    // A frag (16x4 f32): VGPR0/1 = K offsets {2*hi, 2*hi+1}
    v2f a = *(const v2f*)(arow + k + 2 * hi);
    // B frag (4x16 f32): VGPR0/1 = rows k+2*hi, k+2*hi+1; lane = column
    v2f b;
    b.x = X[(size_t)(k + 2 * hi) * BB + b0 + m];
    b.y = X[(size_t)(k + 2 * hi + 1) * BB + b0 + m];
    acc = __builtin_amdgcn_wmma_f32_16x16x4_f32(
        false, a, false, b, (short)0, acc, false, false);
  }
  // C/D layout: VGPR v -> row i0 + v + 8*hi, col b0 + m
  float* yp = Ypart + (size_t)seg * NN * BB;
#pragma unroll
  for (int v = 0; v < 8; ++v)
    yp[(size_t)(i0 + v + 8 * hi) * BB + b0 + m] = acc[v];
}

// ---------------- GEMM: Y[seg][j][b] = sum_{k in seg} A[k][j] * X[k][b] (A transposed) ----
__global__ void emd_gemm_tn(const float* __restrict__ A, const float* __restrict__ X,
                            float* __restrict__ Ypart) {
  const int lane = threadIdx.x & 31;
  const int wv   = blockIdx.x * (blockDim.x >> 5) + (threadIdx.x >> 5);
  const int seg  = wv & (KSEG - 1);
  const int ct   = (wv >> 3) & 1;
  const int rt   = wv >> 4;              // output row tile over M (columns of A)
  const int j0   = rt << 4;
  const int b0   = ct << 4;
  const int hi   = lane >> 4;
  const int m    = lane & 15;

  v8f acc = {};
  const int kbeg = seg * KCHUNK;
  const int kend = kbeg + KCHUNK;
  for (int k = kbeg; k < kend; k += 4) {
    // A frag rows are A^T[j0+m][k..] = A[k..][j0+m]; 16 lanes read 64B contiguous
    v2f a;
    a.x = A[(size_t)(k + 2 * hi) * MM + j0 + m];
    a.y = A[(size_t)(k + 2 * hi + 1) * MM + j0 + m];
    v2f b;
    b.x = X[(size_t)(k + 2 * hi) * BB + b0 + m];
    b.y = X[(size_t)(k + 2 * hi + 1) * BB + b0 + m];
    acc = __builtin_amdgcn_wmma_f32_16x16x4_f32(
        false, a, false, b, (short)0, acc, false, false);
  }
  float* yp = Ypart + (size_t)seg * MM * BB;
#pragma unroll
  for (int v = 0; v < 8; ++v)
    yp[(size_t)(j0 + v + 8 * hi) * BB + b0 + m] = acc[v];
}

// ---------------- f update: F = p / (K g); err[b] = max_i |log(new/old)| --------------
__global__ void emd_update_f(const float* __restrict__ p, const float* __restrict__ Spart,
                             float* __restrict__ F, float* __restrict__ err,
                             const int* __restrict__ done) {
  const int b = blockIdx.x;       // 32 blocks
  const int t = threadIdx.x;      // 256 threads
  __shared__ float red[256];
  const int isdone = done[b];
  float lm = 0.0f;
  for (int i = t; i < NN; i += 256) {
    float s = 0.0f;
#pragma unroll
    for (int seg = 0; seg < KSEG; ++seg)
      s += Spart[(size_t)seg * NN * BB + (size_t)i * BB + b];
    float nf = p[(size_t)b * NN + i] / s;
    if (!isdone) {
      float of = F[(size_t)i * BB + b];
      lm = fmaxf(lm, fabsf(__logf(nf / of)));
      F[(size_t)i * BB + b] = nf;
    }
  }
  red[t] = lm;
  __syncthreads();
  for (int off = 128; off > 0; off >>= 1) {
    if (t < off) red[t] = fmaxf(red[t], red[t + off]);
    __syncthreads();
  }
  if (t == 0) err[b] = red[0];
}

// ---------------- g update: G = q / (K^T f); then sticky done |= err < thresh ---------
__global__ void emd_update_g(const float* __restrict__ q, const float* __restrict__ Tpart,
                             float* __restrict__ G, const float* __restrict__ err,
                             int* __restrict__ done) {
  const int b = blockIdx.x;
  const int t = threadIdx.x;
  const int isdone = done[b];     // OLD done gates this iteration's commits
  for (int j = t; j < MM; j += 256) {
    float s = 0.0f;
#pragma unroll
    for (int seg = 0; seg < KSEG; ++seg)
      s += Tpart[(size_t)seg * MM * BB + (size_t)j * BB + b];
    float ng = q[(size_t)b * MM + j] / s;
    if (!isdone) G[(size_t)j * BB + b] = ng;
  }
  __syncthreads();
  if (t == 0 && !isdone && err[b] < THRESH) done[b] = 1;
}

// ---------------- final: dist[b] = sum_i F[i][b] * ((C.*K) G)[i][b] -------------------
__global__ void emd_final_reduce(const float* __restrict__ F, const float* __restrict__ Spart,
                                 float* __restrict__ out) {
  const int b = blockIdx.x;
  const int t = threadIdx.x;
  __shared__ float red[256];
  float acc = 0.0f;
  for (int i = t; i < NN; i += 256) {
    float s = 0.0f;
#pragma unroll
    for (int seg = 0; seg < KSEG; ++seg)
      s += Spart[(size_t)seg * NN * BB + (size_t)i * BB + b];
    acc += F[(size_t)i * BB + b] * s;
  }
  red[t] = acc;
  __syncthreads();
  for (int off = 128; off > 0; off >>= 1) {
    if (t < off) red[t] += red[t + off];
    __syncthreads();
  }
  if (t == 0) out[b] = red[0];
}

extern "C" void kernel_launch(void* const* d_in, const int* in_sizes, int n_in,
                              void* d_out, int out_size, void* d_ws, size_t ws_size,
                              hipStream_t stream) {
  const float* p = (const float*)d_in[0];   // [B, N]
  const float* q = (const float*)d_in[1];   // [B, M]
  const float* C = (const float*)d_in[2];   // [N, M]
  float* out = (float*)d_out;               // [B]

  float* w    = (float*)d_ws;
  float* K    = w;                          // [N][M]
  float* W    = K + NM;                     // [N][M]  (C .* K)
  float* F    = W + NM;                     // [N][32]
  float* G    = F + (size_t)NN * BB;        // [M][32]
  float* SP   = G + (size_t)MM * BB;        // [KSEG][N][32] partials (reused for T)
  float* err  = SP + (size_t)KSEG * NN * BB;
  int*   done = (int*)(err + BB);

  emd_init_kw<<<NM / 256, 256, 0, stream>>>(C, K, W);
  emd_init_fg<<<(NN * BB) / 256, 256, 0, stream>>>(F, G, err, done);

  // 2048 waves per GEMM: 128 row-tiles x 2 col-tiles x 8 k-segments
  const int GEMM_BLOCKS = (128 * 2 * KSEG * 32) / 256;   // 256 blocks of 8 waves
  for (int it = 0; it < N_ITERS; ++it) {
    emd_gemm_nt<<<GEMM_BLOCKS, 256, 0, stream>>>(K, G, SP);     // S = K g
    emd_update_f<<<BB, 256, 0, stream>>>(p, SP, F, err, done);  // f = p/S, err
    emd_gemm_tn<<<GEMM_BLOCKS, 256, 0, stream>>>(K, F, SP);     // T = K^T f
    emd_update_g<<<BB, 256, 0, stream>>>(q, SP, G, err, done);  // g = q/T, done|=
  }

  emd_gemm_nt<<<GEMM_BLOCKS, 256, 0, stream>>>(W, G, SP);       // (C.*K) g
  emd_final_reduce<<<BB, 256, 0, stream>>>(F, SP, out);
}